// PatchwiseFFCDecoder_59785944760416
// MI455X (gfx1250) — compile-verified
//
#include <hip/hip_runtime.h>
#include <math.h>

// MI455X / gfx1250, wave32. FFT-as-WMMA: 256-pt FFT = D16*(16x16)reshape,
// twiddle, *D16 via V_WMMA_F32_16X16X4_F32 (fp32 matrix path, 4 WMMA per
// real 16x16x16 matmul). One wave = one transform.

typedef __attribute__((ext_vector_type(2))) float v2f;
typedef __attribute__((ext_vector_type(8))) float v8f;

#define EPSBN 1e-5f
#define NPIX  4194304.0f   // 4 * 1024 * 1024
#define SRE_OFF 0
#define SIM_OFF 33024      // 256*129
#define WSCR_OFF 66048     // + wave*512 (512 floats per-wave scratch)
#define LDS_FLOATS (66048 + 16*512)

__device__ __forceinline__ v8f wmma4(v2f a, v2f b, v8f c) {
#if defined(__HIP_DEVICE_COMPILE__)
  // (neg_a, A, neg_b, B, c_mod, C, reuse_a, reuse_b) -- f32 16x16x4 shape
  return __builtin_amdgcn_wmma_f32_16x16x4_f32(false, a, false, b, (short)0, c,
                                               false, false);
#else
  (void)a; (void)b;
  return c;
#endif
}

__device__ __forceinline__ void dswait() {
#if defined(__HIP_DEVICE_COMPILE__)
  asm volatile("s_wait_dscnt 0x0" ::: "memory");
#endif
}

// One wave computes one 256-point DFT (sign=-1 fwd, +1 inv), unnormalized;
// caller supplies the scale. Data element i lives at p{re,im}[i*stride] (LDS).
// LOAD: 0 = real row from global (imag=0)
//       1 = complex from LDS (strided)
//       2 = Hermitian-extend a 129-bin LDS row (irfft semantics: Im of bins
//           0 and 128 ignored)
// STORE:0 = keep bins 0..128 into a 129-wide LDS row (rfft row stage)
//       1 = full complex, strided, with per-bin 2x2 channel filter + bias
//       2 = full complex, strided, plain
//       3 = real part only into outRow[256] + accumulate sum / sumsq
template <int LOAD, int STORE>
__device__ __forceinline__ void wave_fft256(
    float sign, const float* __restrict__ gsrc, float* pre, float* pim,
    int stride, float* wscr, int lane, float scale,
    float A00, float A01, float A10, float A11, float c0, float c1,
    float* outRow, float* sAcc, float* qAcc) {
  const int n2 = lane & 15;
  const int half = lane >> 4;  // 0 or 1
  const float W16 = sign * 0.39269908169872414f;    // 2*pi/16
  const float W256 = sign * 0.024543692606170259f;  // 2*pi/256

  // DFT-16 factors: serve as A-operand of matmul1 AND B-operand of matmul2
  // (D16 is symmetric and both layouts index element value by (lane&15)*k).
  v2f dre[4], dim[4];
#pragma unroll
  for (int kc = 0; kc < 4; ++kc)
#pragma unroll
    for (int v = 0; v < 2; ++v) {
      int k = 4 * kc + v + 2 * half;
      float s, c;
      __sincosf(W16 * (float)(n2 * k), &s, &c);
      dre[kc][v] = c;
      dim[kc][v] = s;
    }

  // Load input as B operand: B chunk kc holds rows n1 = 4*kc + v + 2*half,
  // column = lane&15; element index i = 16*n1 + n2.
  v2f br[4], bi[4];
#pragma unroll
  for (int kc = 0; kc < 4; ++kc)
#pragma unroll
    for (int v = 0; v < 2; ++v) {
      int n1 = 4 * kc + v + 2 * half;
      int i = 16 * n1 + n2;
      if (LOAD == 0) {
        br[kc][v] = gsrc[i];
        bi[kc][v] = 0.f;
      } else if (LOAD == 1) {
        br[kc][v] = pre[i * stride];
        bi[kc][v] = pim[i * stride];
      } else {
        if (i <= 128) {
          br[kc][v] = pre[i];
          bi[kc][v] = (i == 0 || i == 128) ? 0.f : pim[i];
        } else {
          br[kc][v] = pre[256 - i];
          bi[kc][v] = -pim[256 - i];
        }
      }
    }

  // matmul1: Y = D16 * M   (complex; Mi==0 when LOAD==0)
  v8f t1 = {0.f, 0.f, 0.f, 0.f, 0.f, 0.f, 0.f, 0.f};
  v8f t2 = t1, yi = t1;
#pragma unroll
  for (int kc = 0; kc < 4; ++kc) t1 = wmma4(dre[kc], br[kc], t1);
  if (LOAD != 0) {
#pragma unroll
    for (int kc = 0; kc < 4; ++kc) t2 = wmma4(dim[kc], bi[kc], t2);
  }
  v8f yr = t1 - t2;
#pragma unroll
  for (int kc = 0; kc < 4; ++kc) yi = wmma4(dim[kc], br[kc], yi);
  if (LOAD != 0) {
#pragma unroll
    for (int kc = 0; kc < 4; ++kc) yi = wmma4(dre[kc], bi[kc], yi);
  }

  // Twiddle W256^(k1*n2) elementwise on the C layout, park Y' in LDS so we
  // can re-load it in the A layout for matmul2.
#pragma unroll
  for (int vc = 0; vc < 8; ++vc) {
    int m = vc + 8 * half;  // k1
    float s, c;
    __sincosf(W256 * (float)(m * n2), &s, &c);
    wscr[m * 16 + n2] = yr[vc] * c - yi[vc] * s;
    wscr[256 + m * 16 + n2] = yr[vc] * s + yi[vc] * c;
  }
  dswait();

  v2f ar[4], ai[4];
#pragma unroll
  for (int kc = 0; kc < 4; ++kc)
#pragma unroll
    for (int v = 0; v < 2; ++v) {
      int k = 4 * kc + v + 2 * half;
      ar[kc][v] = wscr[(lane & 15) * 16 + k];
      ai[kc][v] = wscr[256 + (lane & 15) * 16 + k];
    }

  // matmul2: Z = Y' * D16
  v8f u1 = {0.f, 0.f, 0.f, 0.f, 0.f, 0.f, 0.f, 0.f};
  v8f u2 = u1, zi = u1;
#pragma unroll
  for (int kc = 0; kc < 4; ++kc) u1 = wmma4(ar[kc], dre[kc], u1);
#pragma unroll
  for (int kc = 0; kc < 4; ++kc) u2 = wmma4(ai[kc], dim[kc], u2);
  v8f zr = u1 - u2;
  if (STORE != 3) {
#pragma unroll
    for (int kc = 0; kc < 4; ++kc) zi = wmma4(ar[kc], dim[kc], zi);
#pragma unroll
    for (int kc = 0; kc < 4; ++kc) zi = wmma4(ai[kc], dre[kc], zi);
  }

  // Output bin index: i = 16*k2 + k1, k2 = C-col = lane&15, k1 = vc + 8*half.
#pragma unroll
  for (int vc = 0; vc < 8; ++vc) {
    int i = 16 * n2 + vc + 8 * half;
    float zrv = zr[vc] * scale;
    if (STORE == 0) {
      if (i <= 128) {
        pre[i] = zrv;
        pim[i] = zi[vc] * scale;
      }
    } else if (STORE == 1) {
      float ziv = zi[vc] * scale;
      pre[i * stride] = A00 * zrv + A01 * ziv + c0;
      pim[i * stride] = A10 * zrv + A11 * ziv + c1;
    } else if (STORE == 2) {
      pre[i * stride] = zrv;
      pim[i * stride] = zi[vc] * scale;
    } else {
      outRow[i] = zrv;
      *sAcc += zrv;
      *qAcc += zrv * zrv;
    }
  }
}

// K1: one block per 256x256 patch. rfft rows -> (fft col * filter * ifft col,
// fused per column) -> irfft rows; writes sp to d_out, per-patch sum/sumsq.
__global__ __launch_bounds__(512) void k_fft(
    const float* __restrict__ x, const float* __restrict__ pw1,
    const float* __restrict__ pb1, const float* __restrict__ pw2,
    const float* __restrict__ pb2, float* __restrict__ sp,
    float* __restrict__ psum, float* __restrict__ psq) {
  extern __shared__ float lds[];
  float* sre = lds + SRE_OFF;  // [256][129]
  float* sim = lds + SIM_OFF;
  const int tid = threadIdx.x;
  const int wave = tid >> 5;
  const int lane = tid & 31;
  float* wscr = lds + WSCR_OFF + wave * 512;

  // Fold the two 1x1 convs into a single 2x2 map + bias (uniform per bin).
  float A00 = 0.f, A01 = 0.f, A10 = 0.f, A11 = 0.f;
  float c0 = pb2[0], c1 = pb2[1];
#pragma unroll
  for (int m = 0; m < 16; ++m) {
    float w0 = pw2[m], w1 = pw2[16 + m];
    float a = pw1[2 * m], b = pw1[2 * m + 1];
    A00 += w0 * a;
    A01 += w0 * b;
    A10 += w1 * a;
    A11 += w1 * b;
    c0 += w0 * pb1[m];
    c1 += w1 * pb1[m];
  }

  const int p = blockIdx.x;
  const int b = p >> 4, hp = (p >> 2) & 3, wp = p & 3;
  const float* xbase =
      x + (size_t)b * 1048576 + (size_t)hp * 262144 + (size_t)wp * 256;
  float* spbase =
      sp + (size_t)b * 1048576 + (size_t)hp * 262144 + (size_t)wp * 256;

  float dum0 = 0.f, dum1 = 0.f;

  // Stage A: forward row rFFTs (real in, keep bins 0..128), ortho scale later.
  for (int j = 0; j < 16; ++j) {
    int r = wave + 16 * j;
    const float* grow = xbase + (size_t)r * 1024;
    if (j < 15) __builtin_prefetch(xbase + (size_t)(r + 16) * 1024 + lane * 8, 0, 1);
    wave_fft256<0, 0>(-1.f, grow, sre + r * 129, sim + r * 129, 1, wscr, lane,
                      1.f, 0.f, 0.f, 0.f, 0.f, 0.f, 0.f, nullptr, &dum0, &dum1);
  }
  __syncthreads();

  // Stage B: per column: fwd FFT (x 1/256 ortho) -> uniform filter -> inv FFT.
  for (int k2 = wave; k2 <= 128; k2 += 16) {
    wave_fft256<1, 1>(-1.f, nullptr, sre + k2, sim + k2, 129, wscr, lane,
                      0.00390625f, A00, A01, A10, A11, c0, c1, nullptr, &dum0,
                      &dum1);
    dswait();
    wave_fft256<1, 2>(1.f, nullptr, sre + k2, sim + k2, 129, wscr, lane, 1.f,
                      0.f, 0.f, 0.f, 0.f, 0.f, 0.f, nullptr, &dum0, &dum1);
  }
  __syncthreads();

  // Stage C: inverse row irFFTs (x 1/256 ortho), coalesced store + reduce.
  float acc = 0.f, accq = 0.f;
  for (int j = 0; j < 16; ++j) {
    int r = wave + 16 * j;
    wave_fft256<2, 3>(1.f, nullptr, sre + r * 129, sim + r * 129, 1, wscr, lane,
                      0.00390625f, 0.f, 0.f, 0.f, 0.f, 0.f, 0.f, wscr, &acc,
                      &accq);
    dswait();
    float4* g4 = (float4*)(spbase + (size_t)r * 1024);
    const float4* w4 = (const float4*)wscr;
    g4[lane] = w4[lane];
    g4[lane + 32] = w4[lane + 32];
  }

  __syncthreads();
  sre[tid] = acc;
  sre[512 + tid] = accq;
  __syncthreads();
  for (int s = 256; s > 0; s >>= 1) {
    if (tid < s) {
      sre[tid] += sre[tid + s];
      sre[512 + tid] += sre[512 + tid + s];
    }
    __syncthreads();
  }
  if (tid == 0) {
    psum[p] = sre[0];
    psq[p] = sre[512];
  }
}

// K2: fold 64 patch partials -> mean/var of sp; derive per-channel alpha/beta
// (BN1 biases cancel analytically) and the channel-2 mixing weights.
__global__ void k_stats1(const float* __restrict__ psum,
                         const float* __restrict__ psq,
                         const float* __restrict__ fc1w,
                         const float* __restrict__ g1w,
                         const float* __restrict__ bn1g,
                         const float* __restrict__ bn1b,
                         const float* __restrict__ fc2w,
                         const float* __restrict__ g2w,
                         const float* __restrict__ fc2b,
                         const float* __restrict__ g2b, float* __restrict__ P) {
  __shared__ float s[64], q[64];
  int t = threadIdx.x;
  s[t] = psum[t];
  q[t] = psq[t];
  __syncthreads();
  for (int st = 32; st > 0; st >>= 1) {
    if (t < st) {
      s[t] += s[t + st];
      q[t] += q[t + st];
    }
    __syncthreads();
  }
  if (t == 0) {
    float m = s[0] / NPIX;
    float var = q[0] / NPIX - m * m;
    P[25] = m;
#pragma unroll
    for (int c = 0; c < 8; ++c) {
      float w = fc1w[c] + g1w[c];
      P[c] = w * bn1g[c] * rsqrtf(w * w * var + EPSBN);  // alpha_c
      P[8 + c] = bn1b[c];                                // beta_c
      P[16 + c] = fc2w[c] + g2w[c];                      // v_c
    }
    P[24] = fc2b[0] + g2b[0];  // d
  }
}

__device__ __forceinline__ float gfun(float sv, const float lp[26]) {
  float sm = sv - lp[25];
  float acc = lp[24];
#pragma unroll
  for (int c = 0; c < 8; ++c) {
    float t = fmaf(lp[c], sm, lp[8 + c]);
    acc = fmaf(lp[16 + c], fmaxf(t, 0.f), acc);
  }
  return acc;
}

// K3: out2 = g(sp) stats: per-block partial sum/sumsq (deterministic tree).
__global__ __launch_bounds__(256) void k_map_reduce(const float* __restrict__ sp,
                                                    const float* __restrict__ P,
                                                    float* __restrict__ p2s,
                                                    float* __restrict__ p2q) {
  float lp[26];
#pragma unroll
  for (int i = 0; i < 26; ++i) lp[i] = P[i];
  float s = 0.f, q = 0.f;
  size_t base = (size_t)blockIdx.x * 4096 + threadIdx.x;
#pragma unroll
  for (int j = 0; j < 16; ++j) {
    float t = gfun(sp[base + (size_t)j * 256], lp);
    s += t;
    q += t * t;
  }
  __shared__ float ss[256], qq[256];
  ss[threadIdx.x] = s;
  qq[threadIdx.x] = q;
  __syncthreads();
  for (int st = 128; st > 0; st >>= 1) {
    if (threadIdx.x < st) {
      ss[threadIdx.x] += ss[threadIdx.x + st];
      qq[threadIdx.x] += qq[threadIdx.x + st];
    }
    __syncthreads();
  }
  if (threadIdx.x == 0) {
    p2s[blockIdx.x] = ss[0];
    p2q[blockIdx.x] = qq[0];
  }
}

// K4: fold 1024 partials -> BN2 scale/shift.
__global__ void k_stats2(const float* __restrict__ p2s,
                         const float* __restrict__ p2q,
                         const float* __restrict__ bn2g,
                         const float* __restrict__ bn2b, float* __restrict__ P) {
  __shared__ float s[256], q[256];
  int t = threadIdx.x;
  float a = 0.f, b = 0.f;
#pragma unroll
  for (int j = 0; j < 4; ++j) {
    a += p2s[t + 256 * j];
    b += p2q[t + 256 * j];
  }
  s[t] = a;
  q[t] = b;
  __syncthreads();
  for (int st = 128; st > 0; st >>= 1) {
    if (t < st) {
      s[t] += s[t + st];
      q[t] += q[t + st];
    }
    __syncthreads();
  }
  if (t == 0) {
    float m2 = s[0] / NPIX;
    float v2 = q[0] / NPIX - m2 * m2;
    P[26] = m2;
    P[27] = rsqrtf(v2 + EPSBN) * bn2g[0];
    P[28] = bn2b[0];
  }
}

// K5: final elementwise out = relu((g(sp) - m2) * inv2 + b2), in place.
__global__ __launch_bounds__(256) void k_final(float* __restrict__ buf,
                                               const float* __restrict__ P) {
  float lp[29];
#pragma unroll
  for (int i = 0; i < 29; ++i) lp[i] = P[i];
  size_t i0 = ((size_t)blockIdx.x * 256 + threadIdx.x) * 4;
  float4* p4 = (float4*)(buf + i0);
  float4 v = *p4;
  float* e = (float*)&v;
#pragma unroll
  for (int j = 0; j < 4; ++j) {
    float t = gfun(e[j], lp);
    e[j] = fmaxf(fmaf(t - lp[26], lp[27], lp[28]), 0.f);
  }
  *p4 = v;
}

extern "C" void kernel_launch(void* const* d_in, const int* in_sizes, int n_in,
                              void* d_out, int out_size, void* d_ws,
                              size_t ws_size, hipStream_t stream) {
  (void)in_sizes;
  (void)n_in;
  (void)out_size;
  (void)ws_size;
  const float* x = (const float*)d_in[0];
  const float* pw1 = (const float*)d_in[1];
  const float* pb1 = (const float*)d_in[2];
  const float* pw2 = (const float*)d_in[3];
  const float* pb2 = (const float*)d_in[4];
  // d_in[5] (loc1_b) feeds only y[:, :8], which the reference discards.
  const float* fc1w = (const float*)d_in[6];
  // d_in[7] (fc1_b) cancels inside BN1 (channel-affine shift).
  const float* g1w = (const float*)d_in[8];
  // d_in[9] (g1_b) cancels likewise.
  const float* bn1g = (const float*)d_in[10];
  const float* bn1b = (const float*)d_in[11];
  const float* fc2w = (const float*)d_in[12];
  const float* fc2b = (const float*)d_in[13];
  const float* g2w = (const float*)d_in[14];
  const float* g2b = (const float*)d_in[15];
  const float* bn2g = (const float*)d_in[16];
  const float* bn2b = (const float*)d_in[17];

  float* sp = (float*)d_out;  // reuse output buffer for the sp intermediate
  float* ws = (float*)d_ws;
  float* psum = ws;          // 64
  float* psq = ws + 64;      // 64
  float* p2s = ws + 128;     // 1024
  float* p2q = ws + 1152;    // 1024
  float* P = ws + 2176;      // 32 derived params

  size_t shbytes = (size_t)LDS_FLOATS * sizeof(float);  // 296,960 B < 320 KB
  k_fft<<<64, 512, shbytes, stream>>>(x, pw1, pb1, pw2, pb2, sp, psum, psq);
  k_stats1<<<1, 64, 0, stream>>>(psum, psq, fc1w, g1w, bn1g, bn1b, fc2w, g2w,
                                 fc2b, g2b, P);
  k_map_reduce<<<1024, 256, 0, stream>>>(sp, P, p2s, p2q);
  k_stats2<<<1, 256, 0, stream>>>(p2s, p2q, bn2g, bn2b, P);
  k_final<<<4096, 256, 0, stream>>>((float*)d_out, P);
}